// FUNKAN_41721312314118
// MI455X (gfx1250) — compile-verified
//
#include <hip/hip_runtime.h>
#include <hip/hip_bf16.h>
#include <math.h>

#define R_H 6
#define EPSBN 1e-5f
#define BSZ 8
#define CSZ 64
#define NSZ 16384   // 128*128
#define OC  64
#define DT  128     // d-tile per block in the fused kernel
// Pair-interleaved LDS layouts: row pair (2c,2c+1) stored element-interleaved so a
// lane's two K-components are one aligned ds_load_b64. Strides % 64 == 32 so the
// two 16-lane halves of a wave (pair-row p and p+1) land in disjoint bank halves.
#define YPS 288     // y pair-row stride in words   (2*DT + 32)
#define WPS 160     // wtT pair-row stride in words (2*OC + 32)

typedef __attribute__((ext_vector_type(2))) float v2f;
typedef __attribute__((ext_vector_type(8))) float v8f;

// Hermite functions psi_0..psi_5 of the reference (R=6): t = tanh(x)*sqrt(13)
__device__ __forceinline__ void hermite6(float xn, float p[R_H]) {
    const float SQ13 = 3.60555127546398929312f;   // sqrt(2R+1)
    const float PIQ  = 0.75112554446494248286f;   // pi^(-1/4)
    float t = tanhf(xn) * SQ13;
    float g = __expf(-0.5f * t * t);
    p[0] = PIQ * g;
    p[1] = 1.41421356237309515f * PIQ * t * g;
    p[2] = 1.0f                 * t * p[1] - 0.70710678118654752f * p[0]; // k=2
    p[3] = 0.81649658092772603f * t * p[2] - 0.81649658092772603f * p[1]; // k=3
    p[4] = 0.70710678118654752f * t * p[3] - 0.86602540378443865f * p[2]; // k=4
    p[5] = 0.63245553203367587f * t * p[4] - 0.89442719099991588f * p[3]; // k=5
}

// ---------------- K0: zero atomic accumulators (sum[64], sumsq[64]) -------
__global__ void k_init(float* __restrict__ ws) {
    int i = threadIdx.x;
    if (i < 2 * CSZ) ws[i] = 0.0f;
}

// ---------------- K1: per-channel sum / sumsq over (B,N) ------------------
__global__ void __launch_bounds__(256) k_stats(const float* __restrict__ x,
                                               float* __restrict__ ws) {
    __shared__ float red[256];
    int bc = blockIdx.x;            // 0 .. B*C-1 ; contiguous 16384 floats each
    int c  = bc & (CSZ - 1);
    const float* xp = x + (size_t)bc * NSZ;
    float s = 0.f, s2 = 0.f;
    for (int d = threadIdx.x; d < NSZ; d += 256) {
        float v = xp[d];
        s += v; s2 += v * v;
    }
    red[threadIdx.x] = s; __syncthreads();
    for (int st = 128; st > 0; st >>= 1) {
        if (threadIdx.x < st) red[threadIdx.x] += red[threadIdx.x + st];
        __syncthreads();
    }
    if (threadIdx.x == 0) atomicAdd(&ws[c], red[0]);
    __syncthreads();
    red[threadIdx.x] = s2; __syncthreads();
    for (int st = 128; st > 0; st >>= 1) {
        if (threadIdx.x < st) red[threadIdx.x] += red[threadIdx.x + st];
        __syncthreads();
    }
    if (threadIdx.x == 0) atomicAdd(&ws[CSZ + c], red[0]);
}

// ---------------- K2: fold stats into scale/shift -------------------------
__global__ void k_finalize(const float* __restrict__ gamma,
                           const float* __restrict__ beta,
                           float* __restrict__ ws) {
    int c = threadIdx.x;
    if (c >= CSZ) return;
    const float inv_n = 1.0f / ((float)BSZ * (float)NSZ);
    float mean = ws[c] * inv_n;
    float var  = ws[CSZ + c] * inv_n - mean * mean;
    float rstd = rsqrtf(var + EPSBN);
    float sc   = gamma[c] * rstd;
    ws[2 * CSZ + c] = sc;                       // scale
    ws[3 * CSZ + c] = beta[c] - mean * sc;      // shift
}

// ---------------- K3: cost dot-products + 6-way softmax -> w[b,c,k] ------
__global__ void __launch_bounds__(256) k_cost(const float* __restrict__ x,
                                              float* __restrict__ ws) {
    __shared__ float red[256];
    __shared__ float costk[R_H];
    int bc = blockIdx.x;
    int c  = bc & (CSZ - 1);
    float sc = ws[2 * CSZ + c];
    float sh = ws[3 * CSZ + c];
    const float* xp = x + (size_t)bc * NSZ;

    float acc[R_H];
#pragma unroll
    for (int k = 0; k < R_H; ++k) acc[k] = 0.f;

    for (int d = threadIdx.x; d < NSZ; d += 256) {
        float xn = fmaf(sc, xp[d], sh);
        float p[R_H];
        hermite6(xn, p);
#pragma unroll
        for (int k = 0; k < R_H; ++k) acc[k] = fmaf(p[k], xn, acc[k]);
    }
#pragma unroll
    for (int k = 0; k < R_H; ++k) {
        red[threadIdx.x] = acc[k]; __syncthreads();
        for (int st = 128; st > 0; st >>= 1) {
            if (threadIdx.x < st) red[threadIdx.x] += red[threadIdx.x + st];
            __syncthreads();
        }
        if (threadIdx.x == 0) costk[k] = red[0];
        __syncthreads();
    }
    if (threadIdx.x == 0) {
        float m = costk[0];
#pragma unroll
        for (int k = 1; k < R_H; ++k) m = fmaxf(m, costk[k]);
        float e[R_H], ssum = 0.f;
#pragma unroll
        for (int k = 0; k < R_H; ++k) { e[k] = __expf(costk[k] - m); ssum += e[k]; }
        float inv = 1.0f / ssum;
#pragma unroll
        for (int k = 0; k < R_H; ++k) ws[4 * CSZ + bc * R_H + k] = e[k] * inv;
    }
}

// ---------------- K4: fused y-tile + 64x64 GEMM via V_WMMA_F32_16X16X4_F32
__global__ void __launch_bounds__(256) k_fused(const float* __restrict__ x,
                                               const float* __restrict__ Wt,
                                               const float* __restrict__ bias,
                                               const float* __restrict__ ws,
                                               float* __restrict__ out) {
    __shared__ float y_pair[(CSZ / 2) * YPS];   // y pair-interleaved: 36864 B
    __shared__ float wt_pair[(CSZ / 2) * WPS];  // WtT pair-interleaved: 20480 B
    __shared__ float sc_l[CSZ], sh_l[CSZ], w_l[CSZ * R_H];

    int b      = blockIdx.y;
    int d_base = blockIdx.x * DT;
    int tid    = threadIdx.x;

    // Preload Wt into pair-interleaved transposed layout:
    //   wt_pair[(c>>1)*WPS + o*2 + (c&1)] = Wt[o][c]
    for (int i = tid; i < OC * CSZ; i += 256) {
        int o = i >> 6, c = i & (CSZ - 1);
        wt_pair[(c >> 1) * WPS + o * 2 + (c & 1)] = Wt[i];
    }
    if (tid < CSZ) { sc_l[tid] = ws[2 * CSZ + tid]; sh_l[tid] = ws[3 * CSZ + tid]; }
    for (int i = tid; i < CSZ * R_H; i += 256)
        w_l[i] = ws[4 * CSZ + b * CSZ * R_H + i];
    __syncthreads();

    // Phase 1: y[c][d] tile (64 x 128) -> pair-interleaved LDS; 2 rows/trip.
    int dl   = tid & (DT - 1);
    int cofs = tid >> 7;               // 0 or 1
    for (int cb = 0; cb < CSZ; cb += 2) {
        int c = cb + cofs;
        const float* xrow = &x[(((size_t)b * CSZ + c) * NSZ) + d_base + dl];
        float xv = *xrow;
        if (cb + 2 < CSZ) __builtin_prefetch(xrow + 2 * NSZ, 0, 0); // warm next row
        float xn = fmaf(sc_l[c], xv, sh_l[c]);
        float p[R_H];
        hermite6(xn, p);
        float y = 0.f;
#pragma unroll
        for (int k = 0; k < R_H; ++k) y = fmaf(w_l[c * R_H + k], p[k], y);
        y_pair[(cb >> 1) * YPS + dl * 2 + cofs] = y;   // c>>1 == cb>>1, c&1 == cofs
    }
    __syncthreads();

    // Phase 2: out[b, 0:64, d_base:+128] = Wt(64x64) @ y(64x128) + bias.
    // Wave w owns o-tile (w&3); iterates 4 d-tiles. A fragments hoisted to regs.
    int wave = tid >> 5;
    int lane = tid & 31;
    int half = lane >> 4;              // pair-row +half (K-pair select); M+8 for D
    int lid  = lane & 15;
    int o0   = (wave & 3) << 4;

    v2f afrag[16];                     // A (16x4) fragments for all 16 K-steps
#pragma unroll
    for (int s = 0; s < 16; ++s)       // ka = 4s + 2*half -> pair row 2s + half
        afrag[s] = *(const v2f*)&wt_pair[(2 * s + half) * WPS + (o0 + lid) * 2];

#pragma unroll
    for (int i = 0; i < 4; ++i) {
        int d0 = ((((wave >> 2) << 2) + i) << 4);
        v8f acc = {0.f, 0.f, 0.f, 0.f, 0.f, 0.f, 0.f, 0.f};
#pragma unroll
        for (int s = 0; s < 16; ++s) {
            v2f bb = *(const v2f*)&y_pair[(2 * s + half) * YPS + (d0 + lid) * 2];
            acc = __builtin_amdgcn_wmma_f32_16x16x4_f32(
                false, afrag[s], false, bb, (short)0, acc, false, false);
        }
        int dcol = d_base + d0 + lid;
#pragma unroll
        for (int v = 0; v < 8; ++v) {
            int o = o0 + v + 8 * half;   // D: VGPR v -> M = v + 8*half
            out[(((size_t)b * OC + o) * NSZ) + dcol] = acc[v] + bias[o];
        }
    }
}

extern "C" void kernel_launch(void* const* d_in, const int* in_sizes, int n_in,
                              void* d_out, int out_size, void* d_ws, size_t ws_size,
                              hipStream_t stream) {
    const float* x     = (const float*)d_in[0];
    const float* gamma = (const float*)d_in[1];
    const float* beta  = (const float*)d_in[2];
    const float* Wt    = (const float*)d_in[3];
    const float* bias  = (const float*)d_in[4];
    float* out = (float*)d_out;
    float* ws  = (float*)d_ws;   // [sum 64][sumsq 64][scale 64][shift 64][w 8*64*6]

    hipLaunchKernelGGL(k_init,     dim3(1),             dim3(128), 0, stream, ws);
    hipLaunchKernelGGL(k_stats,    dim3(BSZ * CSZ),     dim3(256), 0, stream, x, ws);
    hipLaunchKernelGGL(k_finalize, dim3(1),             dim3(64),  0, stream, gamma, beta, ws);
    hipLaunchKernelGGL(k_cost,     dim3(BSZ * CSZ),     dim3(256), 0, stream, x, ws);
    hipLaunchKernelGGL(k_fused,    dim3(NSZ / DT, BSZ), dim3(256), 0, stream,
                       x, Wt, bias, ws, out);
}